// NeuronFusedSpecModel_85323820303144
// MI455X (gfx1250) — compile-verified
//
#include <hip/hip_runtime.h>
#include <stdint.h>

// ---------------------------------------------------------------------------
// MI455X (gfx1250) speculative-sampling kernel.
// HBM-bound (~188 MB moved -> ~8us roofline at 23.3 TB/s). No matmul => no
// WMMA; instead we exploit CDNA5's 320KB LDS (whole V=32000 f32 row resident)
// and the async global->LDS DMA path (global_load_async_to_lds_b128 +
// s_wait_asynccnt) so target_probs never transits VGPRs on the way in, and the
// normalization pass is a pure LDS pass (single global read of tp/ti, single
// global write of the output). 2D grid (K+1, B) avoids any integer division.
// ---------------------------------------------------------------------------

typedef float f4 __attribute__((ext_vector_type(4)));
typedef int   i4 __attribute__((ext_vector_type(4)));

#define THREADS 320   // 10 wave32 per workgroup; 32000 = 320*4*25 exactly for V=32000

// ---- CDNA5 async global->LDS copy (ASYNCcnt-tracked, VGPR-free data path) ----
// tp is stream-once data => non-temporal hint keeps L2 for concurrent blocks.
__device__ __forceinline__ void async_load_f4_to_lds(uint32_t lds_byte_addr,
                                                     const float* gptr) {
#if defined(__HIP_DEVICE_COMPILE__)
  // vdst = per-lane LDS byte address, vaddr = 64-bit global address
  asm volatile("global_load_async_to_lds_b128 %0, %1, off th:TH_LOAD_NT"
               :: "v"(lds_byte_addr), "v"(gptr)
               : "memory");
#endif
}

__device__ __forceinline__ void wait_async0() {
#if defined(__HIP_DEVICE_COMPILE__)
#if __has_builtin(__builtin_amdgcn_s_wait_asynccnt)
  __builtin_amdgcn_s_wait_asynccnt(0);
#else
  asm volatile("s_wait_asynccnt 0x0" ::: "memory");
#endif
#endif
}

// Deterministic full-wave (32-lane) butterfly sum via ds_bpermute (no LDS mem).
__device__ __forceinline__ float wave32_sum(float v) {
  int lane = (int)(threadIdx.x & 31);
#pragma unroll
  for (int m = 16; m > 0; m >>= 1) {
    int o = __builtin_amdgcn_ds_bpermute(((lane ^ m) << 2), __float_as_int(v));
    v += __int_as_float(o);
  }
  return v;
}

// ---------------------------------------------------------------------------
// Kernel 1: blockIdx = (j, b). j < K: adjusted-prob row; j == K: straight copy
// of the bonus row. Also emits tp_at_draft[b*K+j] to workspace.
// ---------------------------------------------------------------------------
__global__ void __launch_bounds__(THREADS)
spec_adjust_kernel(const int* __restrict__ draft_ids,
                   const float* __restrict__ draft_probs,
                   const int* __restrict__ target_indices,
                   const float* __restrict__ target_probs,
                   float* __restrict__ out_probs,    // B*(K+1)*V
                   float* __restrict__ tp_at_draft,  // B*K (workspace)
                   int K, int V)
{
  extern __shared__ float smem[];            // V floats row buffer + reduction slots
  const int j   = (int)blockIdx.x;           // 0 .. K
  const int b   = (int)blockIdx.y;           // 0 .. B-1
  const int r   = b * (K + 1) + j;           // flat row id (no division needed)
  const int tid = (int)threadIdx.x;

  const float* tp_row  = target_probs + (size_t)r * V;
  float*       out_row = out_probs    + (size_t)r * V;

  if (j == K) {
    // Bonus position: last_tp passthrough. Stream-once => non-temporal.
    for (int e = tid * 4; e + 3 < V; e += THREADS * 4) {
      f4 v = __builtin_nontemporal_load((const f4*)(tp_row + e));
      __builtin_nontemporal_store(v, (f4*)(out_row + e));
    }
    for (int e = (V & ~3) + tid; e < V; e += THREADS)
      out_row[e] = tp_row[e];
    return;
  }

  const int   did = draft_ids[b * K + j];
  const float dp  = draft_probs[b * K + j];
  const int* ti_row = target_indices + (size_t)r * V;

  // Generic pointer to LDS: low 32 bits are the wave-relative LDS byte address.
  const uint32_t lds_base = (uint32_t)(uintptr_t)(void*)smem;

  // ---- Stage 0: fire-and-forget async DMA of the whole tp row into LDS ----
  // 25 wave-level b128 ops per wave for V=32000 (ASYNCcnt max 63 -> fine).
  for (int e = tid * 4; e + 3 < V; e += THREADS * 4)
    async_load_f4_to_lds(lds_base + (uint32_t)e * 4u, tp_row + e);

  // Kick the first ti chunk into flight *before* draining the async counter,
  // so global ti traffic overlaps the DMA (the wait has a memory clobber).
  const int e0 = tid * 4;
  i4 ti_next = {};
  if (e0 + 3 < V)
    ti_next = __builtin_nontemporal_load((const i4*)(ti_row + e0));

  float s_loc = 0.0f, t_loc = 0.0f;
  wait_async0();   // own wave's LDS region is now valid

  // ---- Stage A: adj = max(tp - dp*eq, 0) computed in-place in LDS ----
  for (int e = e0; e + 3 < V; e += THREADS * 4) {
    i4 ti = ti_next;
    const int en = e + THREADS * 4;
    if (en + 3 < V)
      ti_next = __builtin_nontemporal_load((const i4*)(ti_row + en));
    f4 tp = *(const f4*)(smem + e);                               // ds_load_b128
    f4 adj;
#pragma unroll
    for (int c = 0; c < 4; ++c) {
      float tpe = tp[c];
      bool  eq  = (ti[c] == did);
      t_loc += eq ? tpe : 0.0f;
      float a = fmaxf(tpe - (eq ? dp : 0.0f), 0.0f);
      adj[c] = a;
      s_loc += a;
    }
    *(f4*)(smem + e) = adj;                                       // ds_store_b128
  }
  for (int e = (V & ~3) + tid; e < V; e += THREADS) {   // scalar tail (V % 4)
    float tpe = tp_row[e];
    bool  eq  = (ti_row[e] == did);
    t_loc += eq ? tpe : 0.0f;
    float a = fmaxf(tpe - (eq ? dp : 0.0f), 0.0f);
    smem[e] = a;
    s_loc += a;
  }

  // ---- Stage B: deterministic block reduction of s and tp_at_draft ----
  const int NW = THREADS / 32;
  float* red = smem + V;                    // 2*NW + 2 scratch floats
  s_loc = wave32_sum(s_loc);
  t_loc = wave32_sum(t_loc);
  const int wave = tid >> 5, lane = tid & 31;
  if (lane == 0) { red[wave] = s_loc; red[NW + wave] = t_loc; }
  __syncthreads();
  if (tid == 0) {
    float s = 0.0f, t = 0.0f;
    for (int w = 0; w < NW; ++w) { s += red[w]; t += red[NW + w]; }
    red[2 * NW]     = s;
    red[2 * NW + 1] = t;
    tp_at_draft[b * K + j] = t;
  }
  __syncthreads();

  const float s     = red[2 * NW];
  const bool  zero  = (s < 1e-30f);
  const float scale = zero ? 0.0f : (1.0f / s);  // out = a*scale + addc
  const float addc  = zero ? 1.0f : 0.0f;        // zero-row => all 1.0

  // ---- Stage C: normalize straight out of LDS, non-temporal store ----
  for (int e = e0; e + 3 < V; e += THREADS * 4) {
    f4 a = *(const f4*)(smem + e);
    f4 o;
#pragma unroll
    for (int c = 0; c < 4; ++c) o[c] = fmaf(a[c], scale, addc);
    __builtin_nontemporal_store(o, (f4*)(out_row + e));
  }
  for (int e = (V & ~3) + tid; e < V; e += THREADS)
    out_row[e] = fmaf(smem[e], scale, addc);
}

// ---------------------------------------------------------------------------
// Kernel 2: Threefry-2x32 (exact JAX reproduction, key = (0,1234)), acceptance
// prefix, token/index selection. One thread per batch row.
// ---------------------------------------------------------------------------
__device__ __forceinline__ uint32_t rotl32(uint32_t x, int r) {
  return (x << r) | (x >> (32 - r));
}

__device__ __forceinline__ void threefry2x32(uint32_t k0, uint32_t k1,
                                             uint32_t c0, uint32_t c1,
                                             uint32_t& o0, uint32_t& o1) {
  const uint32_t ks0 = k0, ks1 = k1, ks2 = k0 ^ k1 ^ 0x1BD11BDAu;
  uint32_t x0 = c0 + ks0, x1 = c1 + ks1;
#define TF_R(r) { x0 += x1; x1 = rotl32(x1, (r)); x1 ^= x0; }
  TF_R(13) TF_R(15) TF_R(26) TF_R(6)   x0 += ks1; x1 += ks2 + 1u;
  TF_R(17) TF_R(29) TF_R(16) TF_R(24)  x0 += ks2; x1 += ks0 + 2u;
  TF_R(13) TF_R(15) TF_R(26) TF_R(6)   x0 += ks0; x1 += ks1 + 3u;
  TF_R(17) TF_R(29) TF_R(16) TF_R(24)  x0 += ks1; x1 += ks2 + 4u;
  TF_R(13) TF_R(15) TF_R(26) TF_R(6)   x0 += ks2; x1 += ks0 + 5u;
#undef TF_R
  o0 = x0; o1 = x1;
}

__global__ void spec_select_kernel(const int* __restrict__ draft_ids,
                                   const float* __restrict__ draft_probs,
                                   const int* __restrict__ target_ids,
                                   const float* __restrict__ tp_at_draft,
                                   float* __restrict__ out_tokens,  // B*(K+1)
                                   float* __restrict__ out_index,   // B
                                   int B, int K)
{
  int b = (int)(blockIdx.x * blockDim.x + threadIdx.x);
  if (b >= B) return;

  const int N = B * K;           // even (448) => jax splits counters in half
  const int half = N >> 1;

  bool mask[16];                 // K <= 15
  bool prefix = true;
  int  idx = 0;
  for (int j = 0; j < K; ++j) {
    const int n = b * K + j;
    const uint32_t i = (uint32_t)((n < half) ? n : (n - half));
    uint32_t o0, o1;
    threefry2x32(0u, 1234u, i, i + (uint32_t)half, o0, o1);
    const uint32_t bits = (n < half) ? o0 : o1;
    float u = __uint_as_float((bits >> 9) | 0x3f800000u) - 1.0f;
    u = fmaxf(u, 0.0f);
    const float ratio = fminf(tp_at_draft[n] / draft_probs[n], 1.0f);
    const bool acc = (u < ratio);
    prefix = prefix && acc;
    mask[j] = prefix;
    idx += prefix ? 1 : 0;
  }

  for (int j = 0; j <= K; ++j) {
    int tok = (j < K && mask[j]) ? draft_ids[b * K + j]
                                 : target_ids[b * (K + 1) + j];
    if (j > idx) tok = 0;        // PAD_TOKEN_ID
    out_tokens[b * (K + 1) + j] = (float)tok;
  }
  out_index[b] = (float)idx;
}

// ---------------------------------------------------------------------------
// Host launcher. All dims derived from in_sizes (graph-capture safe).
// ---------------------------------------------------------------------------
extern "C" void kernel_launch(void* const* d_in, const int* in_sizes, int n_in,
                              void* d_out, int out_size, void* d_ws, size_t ws_size,
                              hipStream_t stream)
{
  const int*   draft_ids      = (const int*)  d_in[0];
  const float* draft_probs    = (const float*)d_in[1];
  const int*   target_ids     = (const int*)  d_in[2];
  const int*   target_indices = (const int*)  d_in[3];
  const float* target_probs   = (const float*)d_in[4];
  // d_in[5] is k (device scalar); k == K here, dims derived from sizes:
  const int BK  = in_sizes[0];          // B*K       (draft_ids)
  const int BK1 = in_sizes[2];          // B*(K+1)   (target_ids)
  const int B   = BK1 - BK;
  const int K   = BK / B;
  const int V   = in_sizes[4] / BK1;

  float* out        = (float*)d_out;
  float* out_tokens = out;              // B*(K+1)
  float* out_index  = out + BK1;        // B
  float* out_probs  = out + BK1 + B;    // B*(K+1)*V
  float* ws_tpd     = (float*)d_ws;     // B*K floats of scratch

  const int    NW    = THREADS / 32;
  const size_t shmem = (size_t)(V + 2 * NW + 2) * sizeof(float); // ~128KB < 320KB/WGP

  dim3 grid1((unsigned)(K + 1), (unsigned)B, 1);
  spec_adjust_kernel<<<grid1, THREADS, shmem, stream>>>(
      draft_ids, draft_probs, target_indices, target_probs,
      out_probs, ws_tpd, K, V);

  spec_select_kernel<<<(B + 63) / 64, 64, 0, stream>>>(
      draft_ids, draft_probs, target_ids, ws_tpd,
      out_tokens, out_index, B, K);
}